// Bandsgate_44504451121215
// MI455X (gfx1250) — compile-verified
//
#include <hip/hip_runtime.h>

typedef __attribute__((ext_vector_type(2))) float v2f;
typedef __attribute__((ext_vector_type(8))) float v8f;

#define NB 32
#define NC 25
#define NHID 50
#define HW 256
#define PLANE (HW * HW)   // 65536
#define HO 252            // conv output extent (pad=1, k=7 shrinks 256 -> 252)
#define TILES 64          // 8x8 tiles of 32x32 over 252x252
#define EPSV 1e-8f

__device__ __forceinline__ v8f wmma4(v2f a, v2f b, v8f c) {
  // D = A(16x4, f32) * B(4x16, f32) + C(16x16, f32)
  return __builtin_amdgcn_wmma_f32_16x16x4_f32(false, a, false, b, (short)0, c,
                                               false, false);
}

// ---------------------------------------------------------------------------
// Kernel A: fused dwconv7 + bias + GELU(exact) + dwconv3 + per-tile sum.
// One WG per (plane, 32x32 output tile). x read exactly once from HBM.
// ---------------------------------------------------------------------------
__global__ __launch_bounds__(256) void conv_stage(
    const float* __restrict__ x, const float* __restrict__ w7,
    const float* __restrict__ b7, const float* __restrict__ w3,
    float* __restrict__ partials) {
  __shared__ float sx[40][41];   // input tile + (2 left, 6 right) halo, padded
  __shared__ float sh1[34][35];  // h1 tile + 1 halo for 3x3
  __shared__ float w7s[49];
  __shared__ float w3s[9];
  __shared__ float red[256];

  const int tid = threadIdx.x;
  const int tile = blockIdx.x;    // 0..63
  const int plane = blockIdx.y;   // 0..799
  const int c = plane % NC;
  const int ty0 = (tile >> 3) * 32;
  const int tx0 = (tile & 7) * 32;

  if (tid < 49) w7s[tid] = w7[c * 49 + tid];
  if (tid < 9) w3s[tid] = w3[c * 9 + tid];
  const float bias7 = b7[c];

  const float* xp = x + (size_t)plane * PLANE;
  // Input tile rows ty0-2 .. ty0+37, cols tx0-2 .. tx0+37 (zero padded).
  for (int i = tid; i < 40 * 40; i += 256) {
    int r = i / 40, cc = i - r * 40;
    int gy = ty0 - 2 + r, gx = tx0 - 2 + cc;
    float v = 0.f;
    if (gy >= 0 && gy < HW && gx >= 0 && gx < HW) v = xp[gy * HW + gx];
    sx[r][cc] = v;
  }
  __syncthreads();

  // h1[hy][hx] = sum_{ky,kx} x[hy+ky-1][hx+kx-1]*w7 + b ; then exact GELU.
  for (int i = tid; i < 34 * 34; i += 256) {
    int r = i / 34, cc = i - r * 34;
    int hy = ty0 - 1 + r, hx = tx0 - 1 + cc;
    float o = 0.f;
    if (hy >= 0 && hy < HO && hx >= 0 && hx < HO) {
      float acc = bias7;
#pragma unroll
      for (int ky = 0; ky < 7; ++ky)
#pragma unroll
        for (int kx = 0; kx < 7; ++kx)
          acc = fmaf(sx[r + ky][cc + kx], w7s[ky * 7 + kx], acc);
      o = 0.5f * acc * (1.f + erff(acc * 0.70710678118654752f));
    }
    sh1[r][cc] = o;
  }
  __syncthreads();

  // 3x3 conv on h1, accumulate tile sum only (h2 is never materialized).
  float lsum = 0.f;
  for (int i = tid; i < 32 * 32; i += 256) {
    int r = i >> 5, cc = i & 31;
    int oy = ty0 + r, ox = tx0 + cc;
    if (oy < HO && ox < HO) {
      float acc = 0.f;
#pragma unroll
      for (int ky = 0; ky < 3; ++ky)
#pragma unroll
        for (int kx = 0; kx < 3; ++kx)
          acc = fmaf(sh1[r + ky][cc + kx], w3s[ky * 3 + kx], acc);
      lsum += acc;
    }
  }
  red[tid] = lsum;
  __syncthreads();
  for (int off = 128; off > 0; off >>= 1) {
    if (tid < off) red[tid] += red[tid + off];
    __syncthreads();
  }
  if (tid == 0) partials[plane * TILES + tile] = red[0];
}

// ---------------------------------------------------------------------------
// Kernel B: one wave. Reduce partials -> G(32x25); fc1 (25->50, ReLU) and
// fc2 (50->25, sigmoid) via fp32 WMMA 16x16x4; gate; per-batch sort (top_k).
// ---------------------------------------------------------------------------
__global__ __launch_bounds__(32) void mlp_gate(
    const float* __restrict__ partials, const float* __restrict__ fc1w,
    const float* __restrict__ fc1b, const float* __restrict__ fc2w,
    const float* __restrict__ fc2b, float* __restrict__ wsGate,
    int* __restrict__ idxOut) {
  __shared__ float G[32][28];   // K padded 25 -> 28
  __shared__ float B1[28][64];  // fc1_w^T, N padded 50 -> 64
  __shared__ float O1[32][52];  // relu(fc1 out), K padded 50 -> 52
  __shared__ float B2[52][32];  // fc2_w^T, N padded 25 -> 32
  __shared__ float S[32][25];

  const int lane = threadIdx.x;
  const int nlo = lane & 15;
  const int hi = lane >> 4;  // lanes 16-31 hold K+2/K+3 (A,B) and M+8 (C/D)

  for (int i = lane; i < 32 * 28; i += 32) (&G[0][0])[i] = 0.f;
  for (int i = lane; i < 28 * 64; i += 32) (&B1[0][0])[i] = 0.f;
  for (int i = lane; i < 32 * 52; i += 32) (&O1[0][0])[i] = 0.f;
  for (int i = lane; i < 52 * 32; i += 32) (&B2[0][0])[i] = 0.f;
  __syncthreads();

  const float inv = 1.f / (float)(HO * HO);
  for (int p = lane; p < NB * NC; p += 32) {
    float s = 0.f;
    for (int t = 0; t < TILES; ++t) s += partials[p * TILES + t];
    G[p / NC][p % NC] = s * inv;
  }
  for (int i = lane; i < NHID * NC; i += 32) {  // fc1_w is (50,25) row-major
    int h = i / NC, cc = i - h * NC;
    B1[cc][h] = fc1w[i];
  }
  for (int i = lane; i < NC * NHID; i += 32) {  // fc2_w is (25,50) row-major
    int cc = i / NHID, h = i - cc * NHID;
    B2[h][cc] = fc2w[i];
  }
  __syncthreads();

  // ---- fc1: G(32x25) @ B1(25x50) + b, ReLU.  M tiles:2, N tiles:4, K:7x4
  for (int mt = 0; mt < 2; ++mt)
    for (int nt = 0; nt < 4; ++nt) {
      int col = nt * 16 + nlo;
      float bv = (col < NHID) ? fc1b[col] : 0.f;
      v8f acc;
#pragma unroll
      for (int i = 0; i < 8; ++i) acc[i] = bv;
#pragma unroll
      for (int kk = 0; kk < 7; ++kk) {
        int k0 = kk * 4 + 2 * hi;
        v2f a, bb;
        a.x = G[mt * 16 + nlo][k0];
        a.y = G[mt * 16 + nlo][k0 + 1];
        bb.x = B1[k0][col];
        bb.y = B1[k0 + 1][col];
        acc = wmma4(a, bb, acc);
      }
#pragma unroll
      for (int i = 0; i < 8; ++i) {
        int m = mt * 16 + i + 8 * hi;
        if (col < 52) O1[m][col] = fmaxf(acc[i], 0.f);
      }
    }
  __syncthreads();

  // ---- fc2: O1(32x50) @ B2(50x25) + b, sigmoid.  M:2, N:2, K:13x4
  for (int mt = 0; mt < 2; ++mt)
    for (int nt = 0; nt < 2; ++nt) {
      int col = nt * 16 + nlo;
      float bv = (col < NC) ? fc2b[col] : 0.f;
      v8f acc;
#pragma unroll
      for (int i = 0; i < 8; ++i) acc[i] = bv;
#pragma unroll
      for (int kk = 0; kk < 13; ++kk) {
        int k0 = kk * 4 + 2 * hi;
        v2f a, bb;
        a.x = O1[mt * 16 + nlo][k0];
        a.y = O1[mt * 16 + nlo][k0 + 1];
        bb.x = B2[k0][col];
        bb.y = B2[k0 + 1][col];
        acc = wmma4(a, bb, acc);
      }
#pragma unroll
      for (int i = 0; i < 8; ++i) {
        int m = mt * 16 + i + 8 * hi;
        if (col < NC) S[m][col] = 1.f / (1.f + expf(-acc[i]));
      }
    }
  __syncthreads();

  // ---- gate + per-batch descending sort of 25 values (ties: lower index).
  {
    int b = lane;  // 32 lanes == 32 batches
    float vals[NC];
    int ids[NC];
    for (int c2 = 0; c2 < NC; ++c2) {
      float s = S[b][c2];
      vals[c2] = s;
      ids[c2] = c2;
      wsGate[b * NC + c2] = (s * s) / (s * s + EPSV);
    }
    for (int j = 0; j < NC; ++j) {
      int mi = j;
      for (int k = j + 1; k < NC; ++k)
        if (vals[k] > vals[mi] ||
            (vals[k] == vals[mi] && ids[k] < ids[mi]))
          mi = k;
      float tv = vals[j]; vals[j] = vals[mi]; vals[mi] = tv;
      int ti = ids[j];  ids[j] = ids[mi];  ids[mi] = ti;
      idxOut[b * NC + j] = ids[j];
    }
  }
}

// ---------------------------------------------------------------------------
// Kernel C: gathered gated copy. One WG per (b, rank j, 1024-float chunk).
// float4 (b128) streaming at HBM rate.
// ---------------------------------------------------------------------------
__global__ __launch_bounds__(256) void gather_gate(
    const float* __restrict__ x, const float* __restrict__ wsGate,
    const int* __restrict__ idx, float* __restrict__ out) {
  int t = blockIdx.x;
  int chunk = t & 63;
  int bj = t >> 6;
  int b = bj / NC, j = bj - b * NC;
  int src = idx[b * NC + j];
  float g = wsGate[b * NC + src];
  size_t pixo = (size_t)chunk * 1024 + (size_t)threadIdx.x * 4;
  const float4* in =
      (const float4*)(x + (size_t)(b * NC + src) * PLANE + pixo);
  float4 v = *in;
  v.x *= g; v.y *= g; v.z *= g; v.w *= g;
  size_t obase =
      (j < 10) ? (size_t)(b * 10 + j) * PLANE
               : (size_t)NB * 10 * PLANE + (size_t)(b * 15 + (j - 10)) * PLANE;
  *(float4*)(out + obase + pixo) = v;
}

extern "C" void kernel_launch(void* const* d_in, const int* in_sizes, int n_in,
                              void* d_out, int out_size, void* d_ws,
                              size_t ws_size, hipStream_t stream) {
  const float* x    = (const float*)d_in[0];
  const float* w7   = (const float*)d_in[1];
  const float* b7   = (const float*)d_in[2];
  const float* w3   = (const float*)d_in[3];
  const float* fc1w = (const float*)d_in[4];
  const float* fc1b = (const float*)d_in[5];
  const float* fc2w = (const float*)d_in[6];
  const float* fc2b = (const float*)d_in[7];
  float* out = (float*)d_out;

  float* partials = (float*)d_ws;              // 800*64 floats
  float* wsGate = partials + NB * NC * TILES;  // 800 floats
  // idx output lives at the tail of d_out (after selected + plug tensors).
  int* idxOut = (int*)(out + (size_t)NB * NC * PLANE);

  dim3 gA(TILES, NB * NC);
  conv_stage<<<gA, 256, 0, stream>>>(x, w7, b7, w3, partials);
  mlp_gate<<<1, 32, 0, stream>>>(partials, fc1w, fc1b, fc2w, fc2b, wsGate,
                                 idxOut);
  gather_gate<<<NB * NC * TILES, 256, 0, stream>>>(x, wsGate, idxOut, out);
}